// SharedNet_85985245266580
// MI455X (gfx1250) — compile-verified
//
#include <hip/hip_runtime.h>
#include <hip/hip_bf16.h>

#define DEV __device__ __forceinline__

// Problem geometry (from setup_inputs; compile-time constants)
constexpr int B    = 4;
constexpr int C    = 32;
constexpr int H0   = 64,  W0 = 128;
constexpr int H1   = 128, W1 = 256;
constexpr int IMGH = 512, IMGW = 1024;
constexpr int D0   = 12;     // maxdisp0
constexpr int KS   = 5;      // 2*maxdisp1 - 1

typedef float v2f __attribute__((ext_vector_type(2)));
typedef float v8f __attribute__((ext_vector_type(8)));

// ---------------------------------------------------------------------------
// CDNA5 async global->LDS copy (ASYNCcnt-tracked), GVS addressing:
//   copies 16 bytes from the global base in saddr plus the byte offset in vaddr to the LDS address in vdst
// ---------------------------------------------------------------------------
DEV void async_copy_b128(unsigned lds_byte_addr, unsigned long long gbase,
                         unsigned gbyte_off) {
  asm volatile("global_load_async_to_lds_b128 %0, %1, %2"
               :
               : "v"(lds_byte_addr), "v"(gbyte_off), "s"(gbase)
               : "memory");
}
DEV void async_wait_all() {
  asm volatile("s_wait_asynccnt 0" ::: "memory");
}

// ---------------------------------------------------------------------------
// Kernel 1: scale-0 cost volume + softmax regression
// grid = B*H0 blocks, block = W0 threads (4 waves). LDS: fr0 row [C][W0] 16KB
// ---------------------------------------------------------------------------
__global__ void __launch_bounds__(W0)
k_cost0_regress(const float* __restrict__ fl0, const float* __restrict__ fr0,
                float* __restrict__ plow0) {
  __shared__ __align__(16) float sfr[C * W0];

  const int bh = blockIdx.x;
  const int b = bh / H0, h = bh % H0;
  const int t = threadIdx.x;

  const unsigned lds_base = (unsigned)(size_t)(void*)&sfr[0];
  const unsigned long long gr = (unsigned long long)(size_t)fr0;
#pragma unroll
  for (int it = 0; it < (C * W0) / (W0 * 4); ++it) {   // 8 issues, 16B/lane
    const int idx = (it * W0 + t) * 4;                 // word index in [C*W0)
    const int c = idx >> 7;                            // /W0
    const int w = idx & (W0 - 1);
    const unsigned goff = (unsigned)((((b * C + c) * H0 + h) * W0 + w) * 4);
    async_copy_b128(lds_base + (unsigned)(idx * 4), gr, goff);
  }
  async_wait_all();
  __syncthreads();

  const int w = t;
  float sad[D0];
#pragma unroll
  for (int d = 0; d < D0; ++d) sad[d] = 0.f;

  const float* flcol = fl0 + ((size_t)(b * C) * H0 + h) * W0 + w;
  for (int c = 0; c < C; ++c) {
    const float fv = flcol[(size_t)c * (H0 * W0)];
    const float* rowc = &sfr[c * W0];
#pragma unroll
    for (int d = 0; d < D0; ++d) {
      const float rv = (w >= d) ? rowc[w - d] : 0.f;   // zero where w < d
      sad[d] += fabsf(fv - rv);
    }
  }
  float mn = sad[0];
#pragma unroll
  for (int d = 1; d < D0; ++d) mn = fminf(mn, sad[d]);
  float se = 0.f, swt = 0.f;
#pragma unroll
  for (int d = 0; d < D0; ++d) {
    const float e = __expf(mn - sad[d]);
    se += e;
    swt += e * (float)d;
  }
  plow0[bh * W0 + w] = (swt / se) * ((float)IMGH / (float)H0);
}

// ---------------------------------------------------------------------------
// WMMA-based separable bilinear upsampling (half-pixel centers, edge clamp).
// A 16x16 output tile of an UPx upsample along one axis depends on <=4 (UP=8)
// or <=6 (UP=4) consecutive source lines -> 1 or 2 V_WMMA_F32_16X16X4_F32
// per tile. Operand layouts per CDNA5 ISA (f32 16x16x4):
//   A[m][k]: lanes 0-15 hold K0(v0),K1(v1); lanes 16-31 hold K2(v0),K3(v1)
//   B[k][n]: lanes 0-15 hold K0(v0),K1(v1); lanes 16-31 hold K2(v0),K3(v1)
//   D[j]   : lanes 0-15 -> row j, col=lane; lanes 16-31 -> row j+8, col=lane-16
// One wave = one 16x16 tile; 8 waves per block. EXEC all-ones, no divergence.
// Edge clamping is applied to the *data* indices (clamped 2-tap == jax's
// renormalized triangle at edges for upsampling).
// ---------------------------------------------------------------------------

// weight of source-tap (base+k) for an output coordinate with frac f:
DEV float tap_weight(int r0_rel, float f, int k) {
  // taps are r0_rel (weight 1-f) and r0_rel+1 (weight f)
  return (k == r0_rel) ? (1.f - f) : ((k == r0_rel + 1) ? f : 0.f);
}

// Vertical pass: dst[HIN*UP][W] = upsample_rows(src[HIN][W]); A=weights, B=data
template<int UP, int HIN, int W>
__global__ void __launch_bounds__(256)
k_vup_wmma(const float* __restrict__ src, float* __restrict__ dst) {
  constexpr int HOUT = HIN * UP;
  constexpr int TY = HOUT / 16, TX = W / 16;
  constexpr int NW = (UP == 8) ? 1 : 2;
  const int lane = threadIdx.x & 31;
  const int tile = blockIdx.x * 8 + (threadIdx.x >> 5);  // over B*TY*TX
  const int txi = tile % TX;
  const int tyi = (tile / TX) % TY;
  const int b   = tile / (TX * TY);
  const float* S = src + (size_t)b * HIN * W;
  float* Dp      = dst + (size_t)b * HOUT * W;

  const float inv = 1.0f / (float)UP;
  const int y0 = tyi * 16;
  const int base = (int)floorf(((float)y0 + 0.5f) * inv - 0.5f);
  const int m = lane & 15;
  const int khalf = (lane >> 4) * 2;                 // 0 or 2
  const int col = txi * 16 + m;                      // data/output column

  const float sy = ((float)(y0 + m) + 0.5f) * inv - 0.5f;
  const float r0f = floorf(sy);
  const float f = sy - r0f;
  const int r0_rel = (int)r0f - base;

  v8f acc = {};
#pragma unroll
  for (int it = 0; it < NW; ++it) {
    const int k0 = it * 4 + khalf, k1 = k0 + 1;
    v2f a, dta;
    a.x = tap_weight(r0_rel, f, k0);
    a.y = tap_weight(r0_rel, f, k1);
    const int rA = min(max(base + k0, 0), HIN - 1);  // clamped source rows
    const int rB = min(max(base + k1, 0), HIN - 1);
    dta.x = S[(size_t)rA * W + col];
    dta.y = S[(size_t)rB * W + col];
    acc = __builtin_amdgcn_wmma_f32_16x16x4_f32(false, a, false, dta,
                                                (short)0, acc, false, false);
  }
  const int radd = (lane >> 4) * 8;
#pragma unroll
  for (int j = 0; j < 8; ++j)
    Dp[(size_t)(y0 + j + radd) * W + col] = acc[j];
}

// Horizontal pass: dst[HOUT][WIN*UP] = upsample_cols(src[HOUT][WIN]);
// A=data, B=weights. If ADD, dst = result + addsrc (same shape as dst).
template<int UP, int WIN, int HOUT, bool ADD>
__global__ void __launch_bounds__(256)
k_hup_wmma(const float* __restrict__ src, const float* __restrict__ addsrc,
           float* __restrict__ dst) {
  constexpr int WOUT = WIN * UP;
  constexpr int TY = HOUT / 16, TX = WOUT / 16;
  constexpr int NW = (UP == 8) ? 1 : 2;
  const int lane = threadIdx.x & 31;
  const int tile = blockIdx.x * 8 + (threadIdx.x >> 5);  // over B*TY*TX
  const int txi = tile % TX;
  const int tyi = (tile / TX) % TY;
  const int b   = tile / (TX * TY);
  const float* S = src + (size_t)b * HOUT * WIN;
  float* Dp      = dst + (size_t)b * HOUT * WOUT;

  const float inv = 1.0f / (float)UP;
  const int x0 = txi * 16;
  const int y0 = tyi * 16;
  const int base = (int)floorf(((float)x0 + 0.5f) * inv - 0.5f);
  const int m = lane & 15;                           // A row / D,B column idx
  const int khalf = (lane >> 4) * 2;

  const float sx = ((float)(x0 + m) + 0.5f) * inv - 0.5f;
  const float r0f = floorf(sx);
  const float f = sx - r0f;
  const int r0_rel = (int)r0f - base;

  v8f acc = {};
#pragma unroll
  for (int it = 0; it < NW; ++it) {
    const int k0 = it * 4 + khalf, k1 = k0 + 1;
    v2f a, wt;
    const int cA = min(max(base + k0, 0), WIN - 1);  // clamped source cols
    const int cB = min(max(base + k1, 0), WIN - 1);
    a.x = S[(size_t)(y0 + m) * WIN + cA];            // A[m][k] data
    a.y = S[(size_t)(y0 + m) * WIN + cB];
    wt.x = tap_weight(r0_rel, f, k0);                // B[k][n] weights
    wt.y = tap_weight(r0_rel, f, k1);
    acc = __builtin_amdgcn_wmma_f32_16x16x4_f32(false, a, false, wt,
                                                (short)0, acc, false, false);
  }
  const int radd = (lane >> 4) * 8;
#pragma unroll
  for (int j = 0; j < 8; ++j) {
    const size_t o = (size_t)(y0 + j + radd) * WOUT + x0 + m;
    float v = acc[j];
    if (ADD) v += addsrc[(size_t)b * HOUT * WOUT + o];
    Dp[o] = v;
  }
}

// ---------------------------------------------------------------------------
// Kernel 3: wflow = resize(pred0 -> [H1,W1]) * (H1/IMGH)
// Downscale by 4: jax antialiased triangle, radius 4, 8 taps per dim,
// per-dim weight renormalization at edges. (Not matrix-friendly: keep scalar.)
// ---------------------------------------------------------------------------
__global__ void __launch_bounds__(256)
k_downsample_flow(const float* __restrict__ pred0, float* __restrict__ wflow) {
  const int i = blockIdx.x * blockDim.x + threadIdx.x;   // B*H1*W1 grid
  const int ox = i % W1;
  const int oy = (i / W1) % H1;
  const int b = i / (W1 * H1);
  const float* src = pred0 + (size_t)b * IMGH * IMGW;

  const float scale = 4.f, inv = 0.25f;
  const float cy = ((float)oy + 0.5f) * scale - 0.5f;
  const float cx = ((float)ox + 0.5f) * scale - 0.5f;
  const int ylo = (int)ceilf(cy - scale);
  const int xlo = (int)ceilf(cx - scale);

  float wy[8], wx[8];
  int yc[8], xc[8];
  float sy = 0.f, sx = 0.f;
#pragma unroll
  for (int j = 0; j < 8; ++j) {
    const int y = ylo + j;
    float wv = 1.f - fabsf((float)y - cy) * inv;
    if (y < 0 || y >= IMGH || wv <= 0.f) wv = 0.f;
    wy[j] = wv; sy += wv;
    yc[j] = min(max(y, 0), IMGH - 1);
    const int x = xlo + j;
    float wu = 1.f - fabsf((float)x - cx) * inv;
    if (x < 0 || x >= IMGW || wu <= 0.f) wu = 0.f;
    wx[j] = wu; sx += wu;
    xc[j] = min(max(x, 0), IMGW - 1);
  }
  float acc = 0.f;
#pragma unroll
  for (int jy = 0; jy < 8; ++jy) {
    if (wy[jy] == 0.f) continue;
    const float* row = src + yc[jy] * IMGW;
    float racc = 0.f;
#pragma unroll
    for (int jx = 0; jx < 8; ++jx) racc += wx[jx] * row[xc[jx]];
    acc += wy[jy] * racc;
  }
  wflow[i] = (acc / (sy * sx)) * ((float)H1 / (float)IMGH);
}

// ---------------------------------------------------------------------------
// Kernel 4: scale-1 residual cost volume (5 shifts, bilinear warp) + regress
// grid = B*H1 blocks, block = W1 threads (8 waves). LDS: fr1 row 32KB
// ---------------------------------------------------------------------------
__global__ void __launch_bounds__(W1)
k_cost1_regress(const float* __restrict__ fl1, const float* __restrict__ fr1,
                const float* __restrict__ wflow, float* __restrict__ plow1) {
  __shared__ __align__(16) float sfr[C * W1];

  const int bh = blockIdx.x;
  const int b = bh / H1, h = bh % H1;
  const int t = threadIdx.x;

  const unsigned lds_base = (unsigned)(size_t)(void*)&sfr[0];
  const unsigned long long gr = (unsigned long long)(size_t)fr1;
#pragma unroll
  for (int it = 0; it < (C * W1) / (W1 * 4); ++it) {   // 8 issues, 16B/lane
    const int idx = (it * W1 + t) * 4;
    const int c = idx >> 8;                            // /W1
    const int w = idx & (W1 - 1);
    const unsigned goff = (unsigned)((((b * C + c) * H1 + h) * W1 + w) * 4);
    async_copy_b128(lds_base + (unsigned)(idx * 4), gr, goff);
  }
  async_wait_all();
  __syncthreads();

  const int w = t;
  const float disp = wflow[bh * W1 + w];

  int x0c[KS], x1c[KS];
  float cl[KS], cr[KS];
#pragma unroll
  for (int k = 0; k < KS; ++k) {
    // shift = k-2 ; bd = disp - shift ; xs = w - bd = w - disp + shift
    const float xs = (float)w - disp + (float)(k - 2);
    const float x0f = floorf(xs);
    const float w1 = xs - x0f;
    const int x0 = (int)x0f, x1 = x0 + 1;
    const bool v0 = (x0 >= 0) && (x0 < W1);
    const bool v1 = (x1 >= 0) && (x1 < W1);
    x0c[k] = min(max(x0, 0), W1 - 1);
    x1c[k] = min(max(x1, 0), W1 - 1);
    cl[k] = v0 ? (1.f - w1) : 0.f;
    cr[k] = v1 ? w1 : 0.f;
  }

  float sad[KS];
#pragma unroll
  for (int k = 0; k < KS; ++k) sad[k] = 0.f;

  const float* flcol = fl1 + ((size_t)(b * C) * H1 + h) * W1 + w;
  for (int c = 0; c < C; ++c) {
    const float fv = flcol[(size_t)c * (H1 * W1)];
    const float* rowc = &sfr[c * W1];
#pragma unroll
    for (int k = 0; k < KS; ++k) {
      const float wv = rowc[x0c[k]] * cl[k] + rowc[x1c[k]] * cr[k];
      sad[k] += fabsf(fv - wv);
    }
  }
  float mn = sad[0];
#pragma unroll
  for (int k = 1; k < KS; ++k) mn = fminf(mn, sad[k]);
  float se = 0.f, swt = 0.f;
#pragma unroll
  for (int k = 0; k < KS; ++k) {
    const float e = __expf(mn - sad[k]);
    se += e;
    swt += e * (float)(k - 2);
  }
  plow1[bh * W1 + w] = (swt / se) * ((float)IMGH / (float)H1);
}

// ---------------------------------------------------------------------------
extern "C" void kernel_launch(void* const* d_in, const int* in_sizes, int n_in,
                              void* d_out, int out_size, void* d_ws, size_t ws_size,
                              hipStream_t stream) {
  (void)in_sizes; (void)n_in; (void)out_size; (void)ws_size;

  const float* fl0 = (const float*)d_in[0];
  const float* fr0 = (const float*)d_in[1];
  const float* fl1 = (const float*)d_in[2];
  const float* fr1 = (const float*)d_in[3];
  // d_in[4..7] are img_h, img_w, maxdisp0, maxdisp1 (static; baked in)

  float* pred0 = (float*)d_out;                       // [B,1,IMGH,IMGW]
  float* pred1 = pred0 + (size_t)B * IMGH * IMGW;     // [B,1,IMGH,IMGW]

  float* ws    = (float*)d_ws;
  float* plow0 = ws;                                  // B*H0*W0
  float* wflow = plow0 + B * H0 * W0;                 // B*H1*W1
  float* plow1 = wflow + B * H1 * W1;                 // B*H1*W1
  float* T     = plow1 + B * H1 * W1;                 // up to B*IMGH*W1

  // scale 0
  k_cost0_regress<<<B * H0, W0, 0, stream>>>(fl0, fr0, plow0);
  k_vup_wmma<8, H0, W0><<<(B * (IMGH / 16) * (W0 / 16)) / 8, 256, 0, stream>>>(plow0, T);
  k_hup_wmma<8, W0, IMGH, false><<<(B * (IMGH / 16) * (IMGW / 16)) / 8, 256, 0, stream>>>(T, nullptr, pred0);

  // coarse flow for scale 1
  k_downsample_flow<<<(B * H1 * W1) / 256, 256, 0, stream>>>(pred0, wflow);

  // scale 1
  k_cost1_regress<<<B * H1, W1, 0, stream>>>(fl1, fr1, wflow, plow1);
  k_vup_wmma<4, H1, W1><<<(B * (IMGH / 16) * (W1 / 16)) / 8, 256, 0, stream>>>(plow1, T);
  k_hup_wmma<4, W1, IMGH, true><<<(B * (IMGH / 16) * (IMGW / 16)) / 8, 256, 0, stream>>>(T, pred0, pred1);
}